// DynamicLinearModel_68496138437119
// MI455X (gfx1250) — compile-verified
//
#include <hip/hip_runtime.h>
#include <math.h>

typedef __attribute__((ext_vector_type(2))) float v2f;
typedef __attribute__((ext_vector_type(4))) float v4f;
typedef __attribute__((ext_vector_type(8))) float v8f;

constexpr int RR = 4096;
constexpr int TT = 2048;
constexpr int XDIM = 8;
constexpr int ZDIM = 8;
constexpr int SEGS = 4;                  // waves per block = t-segments
constexpr int SEG_T = TT / SEGS;         // 512
constexpr int TILES_PER_SEG = SEG_T / 16; // 32

#define WMMA_F32X4(A, B, C) \
  __builtin_amdgcn_wmma_f32_16x16x4_f32(false, (A), false, (B), (short)0, (C), false, false)

__global__ __launch_bounds__(SEGS * 32)
void dlm_kernel(const float* __restrict__ X, const float* __restrict__ Z,
                const float* __restrict__ G, const float* __restrict__ eta,
                const float* __restrict__ zeta, const float* __restrict__ gamma,
                float* __restrict__ out)
{
  __shared__ float lds_u[SEGS * 16];

  const int tid  = threadIdx.x;
  const int wave = tid >> 5;
  const int lane = tid & 31;
  const int l15  = lane & 15;
  const int half = lane >> 4;
  const int koff = half * 2;               // K sub-offset per ISA A/B f32 layout
  const int rbase = blockIdx.x * 16;
  const int r = rbase + l15;               // N index (run) owned by this lane

  // G_hat = sigmoid(G[r]); powers of g for carry application
  const float g = 1.0f / (1.0f + __expf(-G[r]));
  float gp[8];
  gp[0] = 1.0f;
  #pragma unroll
  for (int k = 1; k < 8; ++k) gp[k] = gp[k - 1] * g;
  const float g8 = gp[7] * g;
  float gseg = g8;                          // g^(8*2^6) = g^512
  #pragma unroll
  for (int i = 0; i < 6; ++i) gseg = gseg * gseg;

  // Loop-invariant B operands (16x4 K-major slices of eta/zeta/gamma rows)
  const v2f be0 = *(const v2f*)(eta   + r * XDIM + koff);
  const v2f be1 = *(const v2f*)(eta   + r * XDIM + 4 + koff);
  const v2f bz0 = *(const v2f*)(zeta  + r * ZDIM + koff);
  const v2f bz1 = *(const v2f*)(zeta  + r * ZDIM + 4 + koff);
  const v2f bg0 = *(const v2f*)(gamma + r * ZDIM + koff);
  const v2f bg1 = *(const v2f*)(gamma + r * ZDIM + 4 + koff);

  const int seg_t0 = wave * SEG_T;

  // ---------------- Phase 1: zero-carry segment scan (b = Z.gamma only) ----
  float th = 0.0f;  // theta at current tile start, replicated across half-pairs
  for (int tile = 0; tile < TILES_PER_SEG; ++tile) {
    const int tbase = seg_t0 + tile * 16;
    const int trow  = tbase + l15;         // M index (time) of this lane's A rows
    const v2f az0 = *(const v2f*)(Z + trow * ZDIM + koff);
    const v2f az1 = *(const v2f*)(Z + trow * ZDIM + 4 + koff);

    v8f cb = {};
    cb = WMMA_F32X4(az0, bg0, cb);
    cb = WMMA_F32X4(az1, bg1, cb);

    float u = 0.0f;                        // zero-carry scan over this lane's 8 ts
    #pragma unroll
    for (int k = 0; k < 8; ++k) u = fmaf(g, u, cb[k]);
    const float u_lo = __shfl(u, l15, 32);
    const float u_hi = __shfl(u, l15 + 16, 32);
    th = fmaf(g8, fmaf(g8, th, u_lo), u_hi);   // advance theta by 16 steps
  }
  if (lane < 16) lds_u[wave * 16 + l15] = th;  // segment end-value (zero carry-in)
  __syncthreads();

  // carry into this segment: theta[seg_t0] = sum_{j<wave} gseg^(wave-1-j) * u_j
  float carry = 0.0f;
  for (int j = 0; j < wave; ++j)
    carry = fmaf(gseg, carry, lds_u[j * 16 + l15]);

  // ---------------- Phase 2: full tiles with carry, write output -----------
  for (int tile = 0; tile < TILES_PER_SEG; ++tile) {
    const int tbase = seg_t0 + tile * 16;
    const int trow  = tbase + l15;
    const v2f ax0 = *(const v2f*)(X + trow * XDIM + koff);
    const v2f ax1 = *(const v2f*)(X + trow * XDIM + 4 + koff);
    const v2f az0 = *(const v2f*)(Z + trow * ZDIM + koff);
    const v2f az1 = *(const v2f*)(Z + trow * ZDIM + 4 + koff);

    v8f cxz = {};                          // x_term + z_term (K=8 each -> 4 WMMAs)
    cxz = WMMA_F32X4(ax0, be0, cxz);
    cxz = WMMA_F32X4(ax1, be1, cxz);
    cxz = WMMA_F32X4(az0, bz0, cxz);
    cxz = WMMA_F32X4(az1, bz1, cxz);
    v8f cb = {};                           // b = Z.gamma (2 WMMAs)
    cb = WMMA_F32X4(az0, bg0, cb);
    cb = WMMA_F32X4(az1, bg1, cb);

    // local zero-carry scan, keeping per-step local theta
    float tloc[8];
    float u = 0.0f;
    #pragma unroll
    for (int k = 0; k < 8; ++k) { tloc[k] = u; u = fmaf(g, u, cb[k]); }
    const float u_lo = __shfl(u, l15, 32);
    const float u_hi = __shfl(u, l15 + 16, 32);
    // carry entering this lane's 8-chunk (high half starts 8 steps later)
    const float carry_half = (lane < 16) ? carry : fmaf(g8, carry, u_lo);

    v4f o0, o1;
    #pragma unroll
    for (int k = 0; k < 4; ++k) o0[k] = cxz[k] + fmaf(gp[k], carry_half, tloc[k]);
    #pragma unroll
    for (int k = 4; k < 8; ++k) o1[k - 4] = cxz[k] + fmaf(gp[k], carry_half, tloc[k]);

    carry = fmaf(g8, fmaf(g8, carry, u_lo), u_hi);   // theta at next tile start

    // lane's 8 outputs are contiguous in t: out[r*T + tbase + 8*half + 0..7]
    float* dst = out + (size_t)r * TT + tbase + half * 8;
    *(v4f*)(dst)     = o0;
    *(v4f*)(dst + 4) = o1;
  }
}

extern "C" void kernel_launch(void* const* d_in, const int* in_sizes, int n_in,
                              void* d_out, int out_size, void* d_ws, size_t ws_size,
                              hipStream_t stream) {
  const float* X     = (const float*)d_in[0];
  const float* Z     = (const float*)d_in[1];
  const float* G     = (const float*)d_in[2];
  const float* eta   = (const float*)d_in[3];
  const float* zeta  = (const float*)d_in[4];
  const float* gamma = (const float*)d_in[5];
  float* out = (float*)d_out;

  dim3 grid(RR / 16);        // 256 blocks, one 16-run tile each
  dim3 block(SEGS * 32);     // 4 wave32s = 4 t-segments
  hipLaunchKernelGGL(dlm_kernel, grid, block, 0, stream,
                     X, Z, G, eta, zeta, gamma, out);
}